// GCN_56581899158114
// MI455X (gfx1250) — compile-verified
//
#include <hip/hip_runtime.h>
#include <hip/hip_bf16.h>

// ---------------------------------------------------------------------------
// GCN forward on MI455X (gfx1250): 3x (scale -> edge scatter-add -> WMMA GEMM)
// fp32 end-to-end; GEMM uses V_WMMA_F32_16X16X4_F32 (native fp32 matrix op).
// ---------------------------------------------------------------------------

#define N_NODES     50000
#define N_EDGES     800000
#define IN_FEATS    128
#define H_FEATS     128
#define NUM_CLASSES 16

typedef __attribute__((ext_vector_type(2))) float v2f;
typedef __attribute__((ext_vector_type(8))) float v8f;

// ---- degree accumulation (float atomics; deg arrays pre-zeroed) -----------
__global__ void gcn_degree_kernel(const int* __restrict__ src,
                                  const int* __restrict__ dst,
                                  float* __restrict__ deg_out,
                                  float* __restrict__ deg_in, int E) {
    int e = blockIdx.x * blockDim.x + threadIdx.x;
    if (e < E) {
        atomicAdd(&deg_out[src[e]], 1.0f);
        atomicAdd(&deg_in[dst[e]], 1.0f);
    }
}

// ---- norm = rsqrt(max(deg,1)) ---------------------------------------------
__global__ void gcn_norm_kernel(const float* __restrict__ deg_out,
                                const float* __restrict__ deg_in,
                                float* __restrict__ norm_out,
                                float* __restrict__ norm_in, int n) {
    int i = blockIdx.x * blockDim.x + threadIdx.x;
    if (i < n) {
        norm_out[i] = __frsqrt_rn(fmaxf(deg_out[i], 1.0f));
        norm_in[i]  = __frsqrt_rn(fmaxf(deg_in[i], 1.0f));
    }
}

// ---- hs = h * norm_out[:,None]  (128 feats, one float4 per thread) --------
__global__ void gcn_scale_kernel(const float* __restrict__ h,
                                 const float* __restrict__ norm_out,
                                 float* __restrict__ hs, int n_nodes) {
    int t = blockIdx.x * blockDim.x + threadIdx.x;   // n_nodes * 32 threads
    int node = t >> 5;
    int c4   = t & 31;
    if (node >= n_nodes) return;
    float s = norm_out[node];
    float4 v = ((const float4*)(h + (size_t)node * 128))[c4];
    v.x *= s; v.y *= s; v.z *= s; v.w *= s;
    ((float4*)(hs + (size_t)node * 128))[c4] = v;
}

// ---- agg[dst] += hs[src]  (32 threads/edge, float4 gather, 4 atomics) -----
__global__ void gcn_aggregate_kernel(const float* __restrict__ hs,
                                     const int* __restrict__ src,
                                     const int* __restrict__ dst,
                                     float* __restrict__ agg, int E) {
    long long t = (long long)blockIdx.x * blockDim.x + threadIdx.x;
    int e = (int)(t >> 5);
    int c = (int)(t & 31) * 4;
    if (e >= E) return;
    int s = src[e];
    int d = dst[e];
    float4 v = *(const float4*)(hs + (size_t)s * 128 + c);
    float* o = agg + (size_t)d * 128 + c;
    atomicAdd(o + 0, v.x);
    atomicAdd(o + 1, v.y);
    atomicAdd(o + 2, v.z);
    atomicAdd(o + 3, v.w);
}

// ---- out = relu?( (norm_in[:,None]*A) @ W + b ) via V_WMMA_F32_16X16X4_F32 -
// Block = 256 threads = 8 waves. Block m handles rows [16m,16m+16); wave w
// handles output cols [16w,16w+16). K = 128 consumed 4 at a time.
// LDS A-tile rows padded to 132 floats to avoid stride-128 bank conflicts.
__global__ __launch_bounds__(256)
void gcn_gemm_wmma_kernel(const float* __restrict__ A,        // [M,128] (agg)
                          const float* __restrict__ norm_in,  // [M]
                          const float* __restrict__ W,        // [128,N]
                          const float* __restrict__ bias,     // [N]
                          float* __restrict__ out,            // [M,N]
                          int N, int relu) {
    const int K = 128;
    __shared__ float As[16 * 132];

    const int wave = threadIdx.x >> 5;
    const int lane = threadIdx.x & 31;
    const int half = lane >> 4;      // which half-wave (selects K pair / M+8)
    const int l16  = lane & 15;      // row (A) / col (B,C)
    const int tileM = blockIdx.x * 16;
    const int tileN = wave * 16;

    // Cooperative load of the 16x128 A tile, fused with norm_in scaling.
    for (int i = threadIdx.x; i < 16 * 32; i += 256) {
        int r  = i >> 5;
        int c4 = i & 31;
        float4 v = ((const float4*)(A + (size_t)(tileM + r) * K))[c4];
        float s = norm_in[tileM + r];
        v.x *= s; v.y *= s; v.z *= s; v.w *= s;
        ((float4*)(As + r * 132))[c4] = v;
    }
    __syncthreads();

    if (tileN >= N) return;          // wave-uniform: EXEC stays all-ones

    v8f acc = {};
    for (int k0 = 0; k0 < K; k0 += 4) {
        int ka = k0 + 2 * half;      // lanes 0-15: K=k0,k0+1 ; 16-31: k0+2,k0+3
        v2f a, b;
        a.x = As[l16 * 132 + ka];
        a.y = As[l16 * 132 + ka + 1];
        b.x = W[(size_t)ka * N + tileN + l16];
        b.y = W[(size_t)(ka + 1) * N + tileN + l16];
        acc = __builtin_amdgcn_wmma_f32_16x16x4_f32(
            /*neg_a=*/false, a, /*neg_b=*/false, b,
            /*c_mod=*/(short)0, acc, /*reuse_a=*/false, /*reuse_b=*/false);
    }

    float bb = bias[tileN + l16];
    #pragma unroll
    for (int r = 0; r < 8; ++r) {
        int row = tileM + r + 8 * half;   // C/D layout: VGPR r -> M=r (+8 hi)
        float v = acc[r] + bb;
        if (relu) v = fmaxf(v, 0.0f);
        out[(size_t)row * N + tileN + l16] = v;
    }
}

// ---------------------------------------------------------------------------
extern "C" void kernel_launch(void* const* d_in, const int* in_sizes, int n_in,
                              void* d_out, int out_size, void* d_ws, size_t ws_size,
                              hipStream_t stream) {
    const float* in_feat = (const float*)d_in[0];
    const int*   e_src   = (const int*)d_in[1];
    const int*   e_dst   = (const int*)d_in[2];
    const float* W1 = (const float*)d_in[3];
    const float* b1 = (const float*)d_in[4];
    const float* W2 = (const float*)d_in[5];
    const float* b2 = (const float*)d_in[6];
    const float* W3 = (const float*)d_in[7];
    const float* b3 = (const float*)d_in[8];
    float* out = (float*)d_out;

    const size_t NF = (size_t)N_NODES * 128;  // 6.4M floats per feature buffer
    float* B0 = (float*)d_ws;                 // scaled features (hs)
    float* B1 = B0 + NF;                      // aggregation accumulator
    float* B2 = B1 + NF;                      // layer output ping-pong
    float* deg_out  = B2 + NF;
    float* deg_in   = deg_out + N_NODES;
    float* norm_out = deg_in + N_NODES;
    float* norm_in  = norm_out + N_NODES;

    const int TB = 256;
    const int edgeBlocks  = (N_EDGES + TB - 1) / TB;                   // 3125
    const int aggBlocks   = (int)(((long long)N_EDGES * 32 + TB - 1) / TB);
    const int scaleBlocks = (N_NODES * 32 + TB - 1) / TB;              // 6250
    const int nodeBlocks  = (N_NODES + TB - 1) / TB;
    const int gemmBlocks  = N_NODES / 16;                              // 3125

    // Degrees + norms (deg arrays are contiguous -> single memset).
    hipMemsetAsync(deg_out, 0, 2 * (size_t)N_NODES * sizeof(float), stream);
    gcn_degree_kernel<<<edgeBlocks, TB, 0, stream>>>(e_src, e_dst, deg_out, deg_in, N_EDGES);
    gcn_norm_kernel<<<nodeBlocks, TB, 0, stream>>>(deg_out, deg_in, norm_out, norm_in, N_NODES);

    // ---- Layer 1: in_feat -> B2 (relu) ----
    gcn_scale_kernel<<<scaleBlocks, TB, 0, stream>>>(in_feat, norm_out, B0, N_NODES);
    hipMemsetAsync(B1, 0, NF * sizeof(float), stream);
    gcn_aggregate_kernel<<<aggBlocks, TB, 0, stream>>>(B0, e_src, e_dst, B1, N_EDGES);
    gcn_gemm_wmma_kernel<<<gemmBlocks, TB, 0, stream>>>(B1, norm_in, W1, b1, B2, H_FEATS, 1);

    // ---- Layer 2: B2 -> B2 (relu) ----
    gcn_scale_kernel<<<scaleBlocks, TB, 0, stream>>>(B2, norm_out, B0, N_NODES);
    hipMemsetAsync(B1, 0, NF * sizeof(float), stream);
    gcn_aggregate_kernel<<<aggBlocks, TB, 0, stream>>>(B0, e_src, e_dst, B1, N_EDGES);
    gcn_gemm_wmma_kernel<<<gemmBlocks, TB, 0, stream>>>(B1, norm_in, W2, b2, B2, H_FEATS, 1);

    // ---- Layer 3: B2 -> out (no relu, N=16) ----
    gcn_scale_kernel<<<scaleBlocks, TB, 0, stream>>>(B2, norm_out, B0, N_NODES);
    hipMemsetAsync(B1, 0, NF * sizeof(float), stream);
    gcn_aggregate_kernel<<<aggBlocks, TB, 0, stream>>>(B0, e_src, e_dst, B1, N_EDGES);
    gcn_gemm_wmma_kernel<<<gemmBlocks, TB, 0, stream>>>(B1, norm_in, W3, b3, out, NUM_CLASSES, 0);
}